// BidiagonalThomasFactorization_56856777064909
// MI455X (gfx1250) — compile-verified
//
#include <hip/hip_runtime.h>

// ---------------------------------------------------------------------------
// Block-bidiagonal Thomas solve for MI455X (gfx1250, wave32).
//   A: (1000, 512, 16, 16) f32   diagonal blocks (diagonally dominant)
//   B: ( 999, 512, 16, 16) f32   sub-diagonal blocks
//   v: (512, 16000)        f32   rhs
//   out: (512, 16000)      f32
//
// One workgroup (8 waves) per batch. Segments of 8 blocks:
//   stage 1 (8 waves in parallel): GJ-invert A_i (registers + ds_bpermute),
//            C_i = Ainv*B_{i-1} via v_wmma_f32_16x16x4_f32, w_i = Ainv*v_i,
//            results to double-buffered LDS.
//   stage 2 (wave 0, pipelined):   y_i = w_i - C_i * y_{i-1}.
// ---------------------------------------------------------------------------

typedef __attribute__((ext_vector_type(2))) float v2f;
typedef __attribute__((ext_vector_type(4))) float f4;
typedef __attribute__((ext_vector_type(8))) float v8f;

#define NBLK 1000
#define SBAT 512
#define SBLK 16
#define SEG  8                 // blocks per segment == waves per workgroup
#define NSEG (NBLK / SEG)

static_assert(NBLK % SEG == 0, "chain must divide into segments");

__device__ __forceinline__ float rdlane(float x, int lane) {
  return __builtin_bit_cast(float, __builtin_amdgcn_readlane(__builtin_bit_cast(int, x), lane));
}
__device__ __forceinline__ float bperm(int idx_bytes, float x) {
  return __builtin_bit_cast(float, __builtin_amdgcn_ds_bpermute(idx_bytes, __builtin_bit_cast(int, x)));
}
__device__ __forceinline__ float swapx1(float x) {  // ds_swizzle SWAPX1 (xor=1, and=0x1f)
  return __builtin_bit_cast(float, __builtin_amdgcn_ds_swizzle(__builtin_bit_cast(int, x), 0x041F));
}

__global__ __launch_bounds__(256)
void thomas_blk16_kernel(const float* __restrict__ A,
                         const float* __restrict__ B,
                         const float* __restrict__ v,
                         float* __restrict__ out) {
  __shared__ __align__(16) float ldsC[2][SEG][SBLK * SBLK];
  __shared__ __align__(16) float ldsW[2][SEG][SBLK];

  const int b    = blockIdx.x;        // batch
  const int tid  = threadIdx.x;
  const int wave = tid >> 5;
  const int l    = tid & 31;          // lane
  const int h    = l & 1;             // column half (row layout: lane = 2r + h)
  const int r    = l >> 1;            // matrix row owned by this lane pair
  const int hB   = l >> 4;            // WMMA K-subgroup (lanes 0-15 vs 16-31)

  const int idxP = 4 * h;             // bpermute base: pivot-row source lane 2k+h
  const int idxA = (l & 15) * 8;      // bpermute base for Ainv -> WMMA-A conversion

  float ylane = 0.0f;                 // wave 0: y_{i-1}[r], replicated in lane pair

  // ---------------- stage 1: GJ invert + WMMA C = Ainv*B --------------------
  auto gj_stage = [&](int s) {
    const int i   = s * SEG + wave;
    const int im1 = (i == 0) ? 0 : (i - 1);
    const float* Ab = A + ((size_t)i * SBAT + b) * 256;
    const float* Bb = B + ((size_t)im1 * SBAT + b) * 256;

    // A row layout: lane l holds A[r][8h..8h+7]  (elements l*8 .. l*8+7)
    f4 a0 = ((const f4*)Ab)[l * 2 + 0];
    f4 a1 = ((const f4*)Ab)[l * 2 + 1];
    float a[8] = {a0.x, a0.y, a0.z, a0.w, a1.x, a1.y, a1.z, a1.w};
    float inv[8];
#pragma unroll
    for (int j = 0; j < 8; ++j) inv[j] = (8 * h + j == r) ? 1.0f : 0.0f;
    float rv   = v[(size_t)b * (NBLK * SBLK) + i * SBLK + r];
    float invd = 0.0f;

    // stream-ahead: prefetch next segment's A block for this wave
    if (i + SEG < NBLK)
      __builtin_prefetch(A + ((size_t)(i + SEG) * SBAT + b) * 256, 0, 1);

    // Gauss-Jordan, no pivoting (A diagonally dominant), deferred row scaling.
#pragma unroll
    for (int k = 0; k < 16; ++k) {
      const int kl = k & 7, kh = k >> 3;
      float t    = a[kl];
      float p    = swapx1(t);
      float colk = (h == kh) ? t : p;            // A[r][k]
      float pivv = rdlane(colk, 2 * k);          // A[k][k]
      float rp   = __builtin_amdgcn_rcpf(pivv);
      bool  isPiv = (r == k);
      float m    = isPiv ? 0.0f : colk * rp;
      invd       = isPiv ? rp : invd;

      // A side: only columns >= k still matter  -> registers j >= k-8
      const int jloA = (k >= 8) ? (k - 8) : 0;
#pragma unroll
      for (int j = jloA; j < 8; ++j) {
        float prow = bperm(8 * k + idxP, a[j]);  // A[k][8h+j] from lane 2k+h
        a[j] = fmaf(-m, prow, a[j]);
      }
      // inverse side: row k has fill only in columns <= k -> registers j <= min(k,7)
      const int jhiI = (k < 8) ? k : 7;
#pragma unroll
      for (int j = 0; j <= jhiI; ++j) {
        float prow = bperm(8 * k + idxP, inv[j]);
        inv[j] = fmaf(-m, prow, inv[j]);
      }
      float rvk = rdlane(rv, 2 * k);
      rv = fmaf(-m, rvk, rv);
    }
#pragma unroll
    for (int j = 0; j < 8; ++j) inv[j] *= invd;  // Ainv rows scaled
    float w = rv * invd;                         // w = Ainv * v_i

    // C = Ainv * B via 4 chained V_WMMA_F32_16X16X4_F32
    v8f acc = {0.f, 0.f, 0.f, 0.f, 0.f, 0.f, 0.f, 0.f};
#pragma unroll
    for (int kc = 0; kc < 4; ++kc) {
      // A-operand: lane l needs Ainv[l%16][4kc + 2*(l/16) + j], j=0,1
      const int e0  = (4 * kc) & 7;
      const int idx = idxA + ((kc >= 2) ? 4 : 0);  // source lane 2m (+1 for cols>=8)
      v2f aop;
#pragma unroll
      for (int j = 0; j < 2; ++j) {
        float lo = bperm(idx, inv[e0 + j]);        // dest half h'=0
        float hi = bperm(idx, inv[e0 + j + 2]);    // dest half h'=1
        aop[j] = hB ? hi : lo;
      }
      // B-operand loaded straight from global in WMMA layout:
      // lane l holds B[4kc + j + 2*(l/16)][l%16]
      v2f bop;
#pragma unroll
      for (int j = 0; j < 2; ++j)
        bop[j] = Bb[(4 * kc + j) * 16 + hB * 32 + (l & 15)];

#if __has_builtin(__builtin_amdgcn_wmma_f32_16x16x4_f32)
      acc = __builtin_amdgcn_wmma_f32_16x16x4_f32(
          /*neg_a=*/false, aop, /*neg_b=*/false, bop,
          /*c_mod=*/(short)0, acc, /*reuse_a=*/false, /*reuse_b=*/false);
#else
      // VALU fallback (outer products via cross-lane gathers)
#pragma unroll
      for (int jj = 0; jj < 4; ++jj) {
        const int kg = 4 * kc + jj;
        float bn = Bb[kg * 16 + (l & 15)];
#pragma unroll
        for (int vv = 0; vv < 8; ++vv) {
          float am = bperm(8 * vv + 64 * hB + 4 * (kg >> 3), inv[kg & 7]);
          acc[vv] = fmaf(am, bn, acc[vv]);
        }
      }
#endif
    }

    // D-layout -> row-major LDS:  acc[vv] = C[vv + 8*hB][l%16]
    float* Cd = &ldsC[s & 1][wave][0];
#pragma unroll
    for (int vv = 0; vv < 8; ++vv)
      Cd[(vv + 8 * hB) * 16 + (l & 15)] = acc[vv];
    if (h == 0) ldsW[s & 1][wave][r] = w;
  };

  // ---------------- stage 2: serial recurrence (wave 0) ---------------------
  auto serial_stage = [&](int s) {
#pragma unroll 1
    for (int blk = 0; blk < SEG; ++blk) {
      const int i = s * SEG + blk;
      const float* Cr = &ldsC[s & 1][blk][0];
      const f4* Cr4 = (const f4*)(Cr + r * 16 + 8 * h);   // 32B-aligned
      f4 c0 = Cr4[0];
      f4 c1 = Cr4[1];
      float cj[8] = {c0.x, c0.y, c0.z, c0.w, c1.x, c1.y, c1.z, c1.w};
      float part = 0.0f;
#pragma unroll
      for (int j = 0; j < 8; ++j) {
        float yb = bperm(64 * h + 8 * j, ylane);  // y_{i-1}[8h+j] from lane 2(8h+j)
        part = fmaf(cj[j], yb, part);
      }
      float tot = part + swapx1(part);            // full dot product for row r
      float wv  = ldsW[s & 1][blk][r];
      float yn  = (i == 0) ? wv : (wv - tot);
      ylane = yn;
      if (h == 0) out[(size_t)b * (NBLK * SBLK) + i * SBLK + r] = yn;
    }
  };

  // ---------------- pipelined main loop -------------------------------------
  gj_stage(0);
  __syncthreads();
#pragma unroll 1
  for (int s = 0; s < NSEG; ++s) {
    if (wave == 0) serial_stage(s);               // reads buf[s&1]
    if (s + 1 < NSEG) gj_stage(s + 1);            // writes buf[(s+1)&1]
    __syncthreads();
  }
}

extern "C" void kernel_launch(void* const* d_in, const int* in_sizes, int n_in,
                              void* d_out, int out_size, void* d_ws, size_t ws_size,
                              hipStream_t stream) {
  const float* A = (const float*)d_in[0];
  const float* B = (const float*)d_in[1];
  const float* v = (const float*)d_in[2];
  float* out = (float*)d_out;
  (void)in_sizes; (void)n_in; (void)out_size; (void)d_ws; (void)ws_size;
  thomas_blk16_kernel<<<dim3(SBAT), dim3(SEG * 32), 0, stream>>>(A, B, v, out);
}